// MacroGNN_86586540687514
// MI455X (gfx1250) — compile-verified
//
#include <hip/hip_runtime.h>

typedef __attribute__((ext_vector_type(2))) float v2f;
typedef __attribute__((ext_vector_type(8))) float v8f;

#define N_NODES 100000
#define N_EDGES 1250000
#define D 64
#define NT32 (N_NODES / 32)     // 3125 exactly: 32-row tiles per wave
#define WSTRIDE 68              // LDS row stride (floats): bank-conflict-free b64 frags

// ---------------------------------------------------------------- zero fill
__global__ __launch_bounds__(256) void zero_f4(float4* __restrict__ p, int n4) {
  int i = blockIdx.x * blockDim.x + threadIdx.x;
  int stride = gridDim.x * blockDim.x;
  float4 z = make_float4(0.f, 0.f, 0.f, 0.f);
  for (; i < n4; i += stride) p[i] = z;
}

// ------------------------------------------------- edge scatter (mean accum)
// 64 threads per edge: lane-consecutive feature mapping -> coalesced 128B
// gathers from x[src] and coalesced atomic bursts into agg[dst]. All traffic
// is L2-resident (x/agg = 25.6MB each << 192MB L2).
__global__ __launch_bounds__(256) void scatter_edges(
    const float* __restrict__ xin, const int* __restrict__ src,
    const int* __restrict__ dst, float* __restrict__ agg,
    float* __restrict__ cnt, int do_cnt) {
  int gid = blockIdx.x * blockDim.x + threadIdx.x;
  int e = gid >> 6;
  int f = gid & (D - 1);
  if (e >= N_EDGES) return;
  int s = src[e];
  int d = dst[e];
  atomicAdd(&agg[d * D + f], xin[s * D + f]);
  if (do_cnt && f == 0) atomicAdd(&cnt[d], 1.0f);
}

// ------------------------------------------------- fused SAGE linear (WMMA)
// out[32-row tile] = act( (agg/cnt) @ Wl + bl + xin @ Wr )
// Per block: stage Wl^T / Wr^T in LDS (shared by 8 waves); per wave: two
// 16-row M-tiles so each ds_load_b64 B-fragment feeds 2 WMMAs.
__global__ __launch_bounds__(256) void sage_linear_wmma(
    const float* __restrict__ agg, const float* __restrict__ cnt,
    const float* __restrict__ xin, const float* __restrict__ Wl,
    const float* __restrict__ bl, const float* __restrict__ Wr,
    float* __restrict__ out, int relu) {
  __shared__ float WlT[D * WSTRIDE];
  __shared__ float WrT[D * WSTRIDE];

  // Cooperative transpose-stage: coalesced global b32 reads, conflict-free
  // LDS writes (bank = (4n + k) & 63 spreads across n).
  for (int i = threadIdx.x; i < D * D; i += 256) {
    int k = i >> 6;
    int n = i & 63;
    WlT[n * WSTRIDE + k] = Wl[i];
    WrT[n * WSTRIDE + k] = Wr[i];
  }
  __syncthreads();

  int wave = threadIdx.x >> 5;
  int lane = threadIdx.x & 31;
  int t32 = blockIdx.x * 8 + wave;
  if (t32 >= NT32) return;  // wave-uniform: EXEC all-ones at every WMMA

  int m   = lane & 15;           // row within M-tile / column within N-tile
  int kb  = (lane >> 4) << 1;    // low lanes: K pair (k0,k0+1); high: (+2,+3)
  int kb2 = kb >> 1;
  int row0 = t32 * 32 + m;       // M-tile 0
  int row1 = row0 + 16;          // M-tile 1

  float ic0 = 1.0f / fmaxf(cnt[row0], 1.0f);
  float ic1 = 1.0f / fmaxf(cnt[row1], 1.0f);

  const v2f* agg0 = (const v2f*)(agg + row0 * D);
  const v2f* agg1 = (const v2f*)(agg + row1 * D);
  const v2f* x0   = (const v2f*)(xin + row0 * D);
  const v2f* x1   = (const v2f*)(xin + row1 * D);

  // Accumulators seeded with bias (same value in all 8 row-slots of a column).
  v8f acc0[4], acc1[4];
#pragma unroll
  for (int nt = 0; nt < 4; ++nt) {
    float b = bl[nt * 16 + m];
#pragma unroll
    for (int v = 0; v < 8; ++v) { acc0[nt][v] = b; acc1[nt][v] = b; }
  }

#pragma unroll
  for (int kk = 0; kk < 16; ++kk) {
    int k = kk * 4 + kb;
    int ki = kk * 2 + kb2;               // v2f index into 64-float row
    v2f Am0 = agg0[ki] * ic0;            // global_load_b64 + pk_mul
    v2f Ax0 = x0[ki];
    v2f Am1 = agg1[ki] * ic1;
    v2f Ax1 = x1[ki];
#pragma unroll
    for (int nt = 0; nt < 4; ++nt) {
      int n = nt * 16 + m;
      v2f Bl = *(const v2f*)&WlT[n * WSTRIDE + k];   // ds_load_b64
      v2f Br = *(const v2f*)&WrT[n * WSTRIDE + k];
      // Interleave the two M-tiles to break WMMA->WMMA RAW chains.
      acc0[nt] = __builtin_amdgcn_wmma_f32_16x16x4_f32(
          false, Am0, false, Bl, (short)0, acc0[nt], false, false);
      acc1[nt] = __builtin_amdgcn_wmma_f32_16x16x4_f32(
          false, Am1, false, Bl, (short)0, acc1[nt], false, false);
      acc0[nt] = __builtin_amdgcn_wmma_f32_16x16x4_f32(
          false, Ax0, false, Br, (short)0, acc0[nt], false, false);
      acc1[nt] = __builtin_amdgcn_wmma_f32_16x16x4_f32(
          false, Ax1, false, Br, (short)0, acc1[nt], false, false);
    }
  }

  // C/D layout: VGPR v -> row M=v (lanes 0-15) or M=v+8 (lanes 16-31), N=lane&15
  int mb0 = t32 * 32 + ((lane < 16) ? 0 : 8);
#pragma unroll
  for (int nt = 0; nt < 4; ++nt) {
    int n = nt * 16 + m;
#pragma unroll
    for (int v = 0; v < 8; ++v) {
      float v0 = acc0[nt][v];
      float v1 = acc1[nt][v];
      if (relu) { v0 = fmaxf(v0, 0.0f); v1 = fmaxf(v1, 0.0f); }
      out[(mb0 + v) * D + n] = v0;
      out[(mb0 + 16 + v) * D + n] = v1;
    }
  }
}

// ---------------------------------------------------------------------------
extern "C" void kernel_launch(void* const* d_in, const int* in_sizes, int n_in,
                              void* d_out, int out_size, void* d_ws,
                              size_t ws_size, hipStream_t stream) {
  const float* x   = (const float*)d_in[0];
  const int* ei    = (const int*)d_in[1];
  const int* src   = ei;
  const int* dst   = ei + N_EDGES;
  const float* Wl1 = (const float*)d_in[2];
  const float* bl1 = (const float*)d_in[3];
  const float* Wr1 = (const float*)d_in[4];
  const float* Wl2 = (const float*)d_in[5];
  const float* bl2 = (const float*)d_in[6];
  const float* Wr2 = (const float*)d_in[7];
  float* out = (float*)d_out;

  char* ws = (char*)d_ws;
  float* agg = (float*)ws;                                     // 25.6 MB
  float* cnt = (float*)(ws + (size_t)N_NODES * D * 4);         // 0.4 MB
  float* h   = (float*)(ws + (size_t)N_NODES * D * 4 + (size_t)N_NODES * 4);

  const int zgrid = 2048;
  const int sgrid = (N_EDGES * D) / 256;   // 312500, exact
  const int ggrid = (NT32 + 7) / 8;        // 391 blocks x 8 waves

  // Layer 1
  zero_f4<<<zgrid, 256, 0, stream>>>((float4*)agg,
                                     (N_NODES * D + N_NODES) / 4);
  scatter_edges<<<sgrid, 256, 0, stream>>>(x, src, dst, agg, cnt, 1);
  sage_linear_wmma<<<ggrid, 256, 0, stream>>>(agg, cnt, x, Wl1, bl1, Wr1, h, 1);

  // Layer 2 (degree counts are layer-invariant: reuse cnt)
  zero_f4<<<zgrid, 256, 0, stream>>>((float4*)agg, (N_NODES * D) / 4);
  scatter_edges<<<sgrid, 256, 0, stream>>>(h, src, dst, agg, cnt, 0);
  sage_linear_wmma<<<ggrid, 256, 0, stream>>>(agg, cnt, h, Wl2, bl2, Wr2, out, 0);
}